// StandardDeviationPooling_45792941310010
// MI455X (gfx1250) — compile-verified
//
#include <hip/hip_runtime.h>
#include <math.h>

typedef float v2f __attribute__((ext_vector_type(2)));
typedef float v8f __attribute__((ext_vector_type(8)));

#define HH   1024
#define WW   1024
#define HO   511
#define WO   511

#define TOX  64          // output cols per workgroup
#define TOY  32          // output rows per workgroup
#define NROWS 66         // input rows staged per tile  (TOY*2 + 2)
#define NQ4   33         // float4 loads per row (132 cols >= 2*TOX+2 pairs*2)
#define PSTRIDE 68       // LDS row stride in floats (66 pairs, padded)

__global__ __launch_bounds__(256) void stdpool_wmma_kernel(
    const float* __restrict__ x, float* __restrict__ out)
{
    __shared__ float ph[NROWS * PSTRIDE];   // pair sums
    __shared__ float pq[NROWS * PSTRIDE];   // pair sums of squares

    const int tid = threadIdx.x;
    const int b   = blockIdx.z;
    const int ox0 = blockIdx.x * TOX;
    const int oy0 = blockIdx.y * TOY;
    const int cx0 = ox0 * 2;     // input col base (multiple of 128 -> 512B aligned)
    const int ry0 = oy0 * 2;     // input row base

    // ---------------- Phase 1: coalesced float4 loads -> pair sums in LDS ----
    const size_t img = (size_t)b * HH * WW;
    for (int idx = tid; idx < NROWS * NQ4; idx += 256) {
        int row = idx / NQ4;
        int q4  = idx - row * NQ4;
        int gr  = ry0 + row;      if (gr > HH - 1) gr = HH - 1;     // clamp: feeds only masked/A=0 terms
        int gc  = cx0 + 4 * q4;   if (gc > WW - 4) gc = WW - 4;
        const float4 v = *(const float4*)(x + img + (size_t)gr * WW + gc);
        const int o = row * PSTRIDE + 2 * q4;
        ph[o + 0] = v.x + v.y;
        ph[o + 1] = v.z + v.w;
        pq[o + 0] = v.x * v.x + v.y * v.y;
        pq[o + 1] = v.z * v.z + v.w * v.w;
    }
    __syncthreads();

    // ---------------- Phase 2: vertical banded reduction via WMMA f32 --------
    const int lane = tid & 31;
    const int wv   = tid >> 5;        // wave 0..7
    const int n    = lane & 15;       // A-row m == B/D column n for both halves
    const int hi   = lane >> 4;       // lane half
    const int tx   = wv & 3;
    const int ty   = wv >> 2;
    const int lyb  = 32 * ty;         // local input-row base of this wave's window
    const int lxb  = 16 * tx;         // local pair / output-x base

    v8f csum = {0.f,0.f,0.f,0.f,0.f,0.f,0.f,0.f};
    v8f csq  = {0.f,0.f,0.f,0.f,0.f,0.f,0.f,0.f};

    const int tm = 2 * n;             // band: A[m,c]=1 iff 2m <= c <= 2m+3
#pragma unroll
    for (int k = 0; k < 9; ++k) {     // K = 36 (rows 34,35 are A=0 padding)
        const int c0 = 4 * k + 2 * hi;
        const int c1 = c0 + 1;

        v2f a;                        // A 16x4 fragment (banded 0/1, exact in f32)
        a.x = (c0 >= tm && c0 <= tm + 3) ? 1.0f : 0.0f;
        a.y = (c1 >= tm && c1 <= tm + 3) ? 1.0f : 0.0f;

        int r0 = lyb + c0; if (r0 > NROWS - 1) r0 = NROWS - 1;  // pad rows: A=0 there
        int r1 = lyb + c1; if (r1 > NROWS - 1) r1 = NROWS - 1;
        const int col = lxb + n;

        v2f bs, bq;                   // B 4x16 fragments: horizontal sums on the fly
        bs.x = ph[r0 * PSTRIDE + col] + ph[r0 * PSTRIDE + col + 1];
        bs.y = ph[r1 * PSTRIDE + col] + ph[r1 * PSTRIDE + col + 1];
        bq.x = pq[r0 * PSTRIDE + col] + pq[r0 * PSTRIDE + col + 1];
        bq.y = pq[r1 * PSTRIDE + col] + pq[r1 * PSTRIDE + col + 1];

        // D = A x B + C   (v_wmma_f32_16x16x4_f32), same A reused for both
        csum = __builtin_amdgcn_wmma_f32_16x16x4_f32(
                   false, a, false, bs, (short)0, csum, false, false);
        csq  = __builtin_amdgcn_wmma_f32_16x16x4_f32(
                   false, a, false, bq, (short)0, csq,  false, false);
    }

    // ---------------- Epilogue: std = sqrt(max(E[x^2]-E[x]^2,0)) -------------
    const int oxg = ox0 + lxb + n;
    const size_t ob = (size_t)b * ((size_t)HO * WO);
#pragma unroll
    for (int i = 0; i < 8; ++i) {
        const int oyg = oy0 + 16 * ty + i + 8 * hi;   // C/D layout: VGPR i -> M=i / M=i+8
        if (oyg < HO && oxg < WO) {
            const float s    = csum[i];
            const float ss   = csq[i];
            const float mean = s * 0.0625f;
            float var        = ss * 0.0625f - mean * mean;
            var = var > 0.0f ? var : 0.0f;
            out[ob + (size_t)oyg * WO + oxg] = sqrtf(var);
        }
    }
}

extern "C" void kernel_launch(void* const* d_in, const int* in_sizes, int n_in,
                              void* d_out, int out_size, void* d_ws, size_t ws_size,
                              hipStream_t stream)
{
    (void)n_in; (void)d_ws; (void)ws_size; (void)out_size;
    const float* x = (const float*)d_in[0];
    float* out     = (float*)d_out;
    const int B    = in_sizes[0] / (HH * WW);   // 64

    dim3 grid((WO + TOX - 1) / TOX,   // 8
              (HO + TOY - 1) / TOY,   // 16
              B);                     // 64
    stdpool_wmma_kernel<<<grid, 256, 0, stream>>>(x, out);
}